// GCN_790273982476
// MI455X (gfx1250) — compile-verified
//
#include <hip/hip_runtime.h>

typedef float v2f __attribute__((ext_vector_type(2)));
typedef float v8f __attribute__((ext_vector_type(8)));

#define NNODES 10000
#define NEDGES 160000

// ---------------------------------------------------------------------------
// Degree / normalization
// ---------------------------------------------------------------------------
__global__ void deg_init_kernel(float* __restrict__ deg, int n) {
  int i = blockIdx.x * blockDim.x + threadIdx.x;
  if (i < n) deg[i] = 1.0f;  // self-loop contributes 1 to every node's degree
}

__global__ void deg_accum_kernel(const int* __restrict__ dst, float* __restrict__ deg, int e) {
  int i = blockIdx.x * blockDim.x + threadIdx.x;
  if (i < e) atomicAdd(&deg[dst[i]], 1.0f);
}

__global__ void dinv_kernel(const float* __restrict__ deg, float* __restrict__ dinv, int n) {
  int i = blockIdx.x * blockDim.x + threadIdx.x;
  if (i < n) dinv[i] = rsqrtf(deg[i]);  // deg >= 1 always (self loops)
}

// ---------------------------------------------------------------------------
// FP32 WMMA GEMM: C[M,N] = A[M,K] * B[K,N]
// blockDim = 128 (4 waves, 2x2). Each wave owns a 32x32 C tile (2x2 WMMA
// accumulators); block covers 64x64. Per k-step of 4: 2 A-frag + 2 B-frag
// loads feed 4 WMMAs (A and B fragments each reused twice).
// N % 64 == 0 and K % 4 == 0 hold for all call sites; M may be ragged:
// A-row addresses are clamped branchlessly (EXEC stays all-ones through
// every WMMA, per ISA requirement) and only the final stores are guarded.
// ---------------------------------------------------------------------------
__global__ __launch_bounds__(128) void wmma_gemm_f32_kernel(
    const float* __restrict__ A, const float* __restrict__ B,
    float* __restrict__ C, int M, int N, int K) {
  const int wave  = threadIdx.x >> 5;
  const int lane  = threadIdx.x & 31;
  const int l16   = lane & 15;
  const int lhalf = lane >> 4;  // 0: lanes 0-15, 1: lanes 16-31

  const int wm = wave >> 1;  // 0..1
  const int wn = wave & 1;   // 0..1
  const int row0 = blockIdx.y * 64 + wm * 32;
  const int col0 = blockIdx.x * 64 + wn * 32;

  // A-fragment (16x4 f32): lanes 0-15 hold K=k+0,k+1 for row M=l16;
  //                        lanes 16-31 hold K=k+2,k+3 for row M=l16.
  const float* arow[2];
#pragma unroll
  for (int mi = 0; mi < 2; ++mi) {
    int r = row0 + mi * 16 + l16;
    r = r < M ? r : M - 1;  // branchless clamp for ragged last block
    arow[mi] = A + (size_t)r * K + lhalf * 2;
  }
  // B-fragment (4x16 f32): VGPR0 = row k (+2 for upper lanes), VGPR1 = next row.
  const float* bcol[2];
#pragma unroll
  for (int ni = 0; ni < 2; ++ni)
    bcol[ni] = B + (size_t)(lhalf * 2) * N + (col0 + ni * 16 + l16);

  v8f acc[2][2] = {};
#pragma unroll 4
  for (int k = 0; k < K; k += 4) {
    v2f a[2], b[2];
#pragma unroll
    for (int mi = 0; mi < 2; ++mi) {
      a[mi].x = arow[mi][k];
      a[mi].y = arow[mi][k + 1];
    }
#pragma unroll
    for (int ni = 0; ni < 2; ++ni) {
      b[ni].x = bcol[ni][(size_t)k * N];
      b[ni].y = bcol[ni][(size_t)(k + 1) * N];
    }
#pragma unroll
    for (int mi = 0; mi < 2; ++mi)
#pragma unroll
      for (int ni = 0; ni < 2; ++ni)
        acc[mi][ni] = __builtin_amdgcn_wmma_f32_16x16x4_f32(
            /*neg_a=*/false, a[mi], /*neg_b=*/false, b[ni],
            /*c_mod=*/(short)0, acc[mi][ni], /*reuse_a=*/false, /*reuse_b=*/false);
  }

  // C/D layout: VGPR r holds row (r + 8*lhalf), column l16.
#pragma unroll
  for (int mi = 0; mi < 2; ++mi) {
#pragma unroll
    for (int r = 0; r < 8; ++r) {
      int row = row0 + mi * 16 + 8 * lhalf + r;
      if (row < M) {
#pragma unroll
        for (int ni = 0; ni < 2; ++ni)
          C[(size_t)row * N + col0 + ni * 16 + l16] = acc[mi][ni][r];
      }
    }
  }
}

// ---------------------------------------------------------------------------
// out[i,f] = h[i,f]*dinv[i]^2 + bias[f]   (self-loop message + bias)
// ---------------------------------------------------------------------------
__global__ void init_out_kernel(const float* __restrict__ h, const float* __restrict__ dinv,
                                const float* __restrict__ bias, float* __restrict__ out,
                                int n, int c) {
  int idx = blockIdx.x * blockDim.x + threadIdx.x;
  if (idx >= n * c) return;
  int i = idx / c;
  int f = idx - i * c;
  float di = dinv[i];
  out[idx] = h[idx] * di * di + bias[f];
}

// ---------------------------------------------------------------------------
// Edge scatter: out[dst] += h[src] * dinv[src]*dinv[dst]   (one block per edge)
// Target rows live in the 192MB L2, so fp32 atomics resolve in-cache.
// ---------------------------------------------------------------------------
__global__ void scatter_kernel(const float* __restrict__ h, const int* __restrict__ src,
                               const int* __restrict__ dst, const float* __restrict__ dinv,
                               float* __restrict__ out, int c) {
  int e = blockIdx.x;
  int s = src[e];
  int d = dst[e];
  float norm = dinv[s] * dinv[d];
  const float* hs = h + (size_t)s * c;
  float* od = out + (size_t)d * c;
  for (int f = threadIdx.x; f < c; f += blockDim.x)
    atomicAdd(&od[f], hs[f] * norm);
}

__global__ void relu_kernel(float* __restrict__ x, int n) {
  int i = blockIdx.x * blockDim.x + threadIdx.x;
  if (i < n) x[i] = fmaxf(x[i], 0.0f);
}

// ---------------------------------------------------------------------------
// Launch
// ---------------------------------------------------------------------------
extern "C" void kernel_launch(void* const* d_in, const int* in_sizes, int n_in,
                              void* d_out, int out_size, void* d_ws, size_t ws_size,
                              hipStream_t stream) {
  const float* x   = (const float*)d_in[0];
  const int*   ei  = (const int*)d_in[1];   // [2, E] flat: row0 = src, row1 = dst
  const float* W1  = (const float*)d_in[2];
  const float* b1  = (const float*)d_in[3];
  const float* W2  = (const float*)d_in[4];
  const float* b2  = (const float*)d_in[5];
  float* out = (float*)d_out;

  const int N = NNODES;
  const int E = NEDGES;
  const int C0 = 512, C1 = 1024, C2 = 512;

  const int* src = ei;
  const int* dst = ei + E;

  // workspace layout
  float* deg  = (float*)d_ws;           // N
  float* dinv = deg + N;                // N
  float* h    = dinv + N;               // N*C1 (reused for layer2's N*C2)
  float* agg1 = h + (size_t)N * C1;     // N*C1

  const int T = 256;
  const int MB = (N + 63) / 64;  // ragged M blocks

  // ---- normalization coefficients ----
  deg_init_kernel<<<(N + T - 1) / T, T, 0, stream>>>(deg, N);
  deg_accum_kernel<<<(E + T - 1) / T, T, 0, stream>>>(dst, deg, E);
  dinv_kernel<<<(N + T - 1) / T, T, 0, stream>>>(deg, dinv, N);

  // ---- layer 1: h = x @ W1 ; agg1 = scatter + b1 ; relu ----
  {
    dim3 grid(C1 / 64, MB);
    wmma_gemm_f32_kernel<<<grid, 128, 0, stream>>>(x, W1, h, N, C1, C0);
    init_out_kernel<<<((N * C1) + T - 1) / T, T, 0, stream>>>(h, dinv, b1, agg1, N, C1);
    scatter_kernel<<<E, T, 0, stream>>>(h, src, dst, dinv, agg1, C1);
    relu_kernel<<<((N * C1) + T - 1) / T, T, 0, stream>>>(agg1, N * C1);
  }

  // ---- layer 2: h = agg1 @ W2 ; out = scatter + b2 ; relu ----
  {
    dim3 grid(C2 / 64, MB);
    wmma_gemm_f32_kernel<<<grid, 128, 0, stream>>>(agg1, W2, h, N, C2, C1);
    init_out_kernel<<<((N * C2) + T - 1) / T, T, 0, stream>>>(h, dinv, b2, out, N, C2);
    scatter_kernel<<<E, T, 0, stream>>>(h, src, dst, dinv, out, C2);
    relu_kernel<<<((N * C2) + T - 1) / T, T, 0, stream>>>(out, N * C2);
  }
}